// CorrespondenceSoftNMS_38465727103422
// MI455X (gfx1250) — compile-verified
//
#include <hip/hip_runtime.h>

typedef __attribute__((ext_vector_type(2))) float v2f;
typedef __attribute__((ext_vector_type(8))) float v8f;

#define NPTS 8192
#define NTILES (NPTS / 16)   // 512 column tiles of 16

// DELTA = 0.1, SIGMA = 0.05
// C1 = (2/DELTA^2) * log2(e)  -> closeness^2 = exp2(C1*dot - 0.5*C1*q_i - 0.5*C1*q_j)
// C2 = log2(e)/SIGMA          -> out = s_i * exp2(-C2 * penalty)
__device__ const float C1 = 288.53900817779270f;
__device__ const float C2 = 28.853900817779268f;

// g[i] = -0.5 * C1 * (|src_i|^2 + |tgt_i|^2)
__global__ void softnms_precompute_g(const float* __restrict__ src,
                                     const float* __restrict__ tgt,
                                     float* __restrict__ g) {
    int i = blockIdx.x * blockDim.x + threadIdx.x;
    if (i < NPTS) {
        float sx = src[3 * i + 0], sy = src[3 * i + 1], sz = src[3 * i + 2];
        float tx = tgt[3 * i + 0], ty = tgt[3 * i + 1], tz = tgt[3 * i + 2];
        float q = sx * sx + sy * sy + sz * sz + tx * tx + ty * ty + tz * tz;
        g[i] = -0.5f * C1 * q;
    }
}

// One wave (32 lanes) per 16-row tile; 8 waves per block; 64 blocks -> 512 waves.
__global__ __launch_bounds__(256)
void softnms_main(const float* __restrict__ src,
                  const float* __restrict__ tgt,
                  const float* __restrict__ scores,
                  const float* __restrict__ g,
                  float* __restrict__ out) {
    const int lane   = threadIdx.x & 31;
    const int wave   = threadIdx.x >> 5;
    const int tile_i = blockIdx.x * 8 + wave;   // [0, 512)
    const int half   = lane >> 4;               // 0: lanes 0-15 (K=0,1) ; 1: lanes 16-31 (K=2,3)
    const int l15    = lane & 15;

    // ---- A operand: rows of this tile. Lane l15 -> point pi; VGPR0=comp kb, VGPR1=comp kb+1 (comp3==0).
    const int pi  = tile_i * 16 + l15;
    const int o0  = 3 * pi + (half ? 2 : 0);    // valid component for VGPR0
    const int o1  = 3 * pi + (half ? 2 : 1);    // valid address; upper half forced to 0 below
    float asx = src[o0], asy = half ? 0.0f : src[o1];
    float atx = tgt[o0], aty = half ? 0.0f : tgt[o1];
    v2f a_s; a_s.x = asx; a_s.y = asy;
    v2f a_t; a_t.x = atx; a_t.y = aty;

    // ---- Per-lane row constants: VGPR r of C/D holds row M = r + 8*half.
    const int rowbase = tile_i * 16 + 8 * half;
    float gi[8], si[8], pen[8];
#pragma unroll
    for (int r = 0; r < 8; ++r) {
        gi[r]  = g[rowbase + r];
        si[r]  = scores[rowbase + r];
        pen[r] = 0.0f;
    }

    // ---- Sweep all 512 column tiles.
#pragma unroll 2
    for (int tj = 0; tj < NTILES; ++tj) {
        const int pj = tj * 16 + l15;
        const int b0 = 3 * pj + (half ? 2 : 0);
        const int b1 = 3 * pj + (half ? 2 : 1);
        float bsx = src[b0], bsy = half ? 0.0f : src[b1];
        float btx = tgt[b0], bty = half ? 0.0f : tgt[b1];
        v2f b_s; b_s.x = bsx; b_s.y = bsy;
        v2f b_t; b_t.x = btx; b_t.y = bty;

        const float gj = g[pj];        // same value across both halves (column j = pj)
        const float sj = scores[pj];

        // dot = src_i.src_j + tgt_i.tgt_j, fused across two chained f32 WMMAs.
        v8f acc = {};
        acc = __builtin_amdgcn_wmma_f32_16x16x4_f32(false, a_s, false, b_s,
                                                    (short)0, acc, false, false);
        acc = __builtin_amdgcn_wmma_f32_16x16x4_f32(false, a_t, false, b_t,
                                                    (short)0, acc, false, false);

#pragma unroll
        for (int r = 0; r < 8; ++r) {
            // closeness^2 = exp2(C1*dot + g_i + g_j)
            float arg = fmaf(C1, acc[r], gi[r] + gj);
            float w   = __builtin_amdgcn_exp2f(arg);
            pen[r]   += (sj > si[r]) ? w : 0.0f;
        }
    }

    // ---- Reduce over the 16 lanes of each half (columns j mod 16).
#pragma unroll
    for (int r = 0; r < 8; ++r) {
        float v = pen[r];
        v += __shfl_xor(v, 1, 32);
        v += __shfl_xor(v, 2, 32);
        v += __shfl_xor(v, 4, 32);
        v += __shfl_xor(v, 8, 32);
        pen[r] = v;
    }

    // Lane 0 holds rows rowbase..rowbase+7 for half 0; lane 16 for half 1.
    if (l15 == 0) {
#pragma unroll
        for (int r = 0; r < 8; ++r) {
            out[rowbase + r] = si[r] * __builtin_amdgcn_exp2f(-C2 * pen[r]);
        }
    }
}

extern "C" void kernel_launch(void* const* d_in, const int* in_sizes, int n_in,
                              void* d_out, int out_size, void* d_ws, size_t ws_size,
                              hipStream_t stream) {
    const float* src    = (const float*)d_in[0];
    const float* tgt    = (const float*)d_in[1];
    const float* scores = (const float*)d_in[2];
    float* out = (float*)d_out;
    float* g   = (float*)d_ws;   // 8192 floats = 32 KB scratch

    softnms_precompute_g<<<NPTS / 256, 256, 0, stream>>>(src, tgt, g);
    softnms_main<<<NTILES / 8, 256, 0, stream>>>(src, tgt, scores, g, out);
}